// NCECriterion_47588237639704
// MI455X (gfx1250) — compile-verified
//
#include <hip/hip_runtime.h>
#include <hip/hip_bf16.h>
#include <stdint.h>

// ---------------- problem constants (from reference) ----------------
#define KCOLS          4096          // K: columns actually used (col 4096 always masked off)
#define KP1            4097          // row stride in memory
#define ROWS_PER_TILE  4            // one TDM tile = 4 rows x 4096 cols = 64 KB
#define BLOCK_THREADS  128          // 4 waves; wave w handles row w of the tile
#define NBLOCKS        512
#define MIN_PROB       0.001f

typedef __attribute__((ext_vector_type(4))) unsigned int v4u;
typedef __attribute__((ext_vector_type(8))) int          v8i;
typedef __attribute__((ext_vector_type(4))) int          v4i;

// Issue one TDM load: tile of `rows` rows x KCOLS fp32, row stride KP1 elements,
// from global byte address `gaddr` into LDS byte offset `lds_off`.
// Descriptor layout per CDNA5 ISA ch.8 (Tensor DMA Descriptor D#).
__device__ __forceinline__ void tdm_load_tile(uint64_t gaddr, uint32_t lds_off, uint32_t rows)
{
    v4u g0;
    g0.x = 1u;                                      // count=1 (valid), user mode, no gather
    g0.y = lds_off;                                 // lds_addr (bytes)
    g0.z = (uint32_t)(gaddr & 0xFFFFFFFFull);       // global_addr[31:0]
    g0.w = (uint32_t)(gaddr >> 32) | 0x80000000u;   // global_addr[56:32] | type=2 ("image")

    v8i g1;
    g1[0] = (int)0x00020000u;                       // workgroup_mask=0, data_size=2 (4B)
    g1[1] = (int)((uint32_t)KCOLS << 16);           // tensor_dim0[15:0]=4096
    g1[2] = (int)(((uint32_t)KCOLS >> 16) | (rows << 16)); // tensor_dim0 hi | tensor_dim1 lo
    g1[3] = (int)((rows >> 16) | ((uint32_t)KCOLS << 16)); // tensor_dim1 hi | tile_dim0=4096
    g1[4] = (int)rows;                              // tile_dim1=rows, tile_dim2=0 (2D tile)
    g1[5] = (int)KP1;                               // tensor_dim0_stride = 4097 elements
    g1[6] = 0;                                      // stride hi (unused, 2D tile)
    g1[7] = 0;

    v4i gz = {0, 0, 0, 0};
#if __clang_major__ >= 23
    v8i gz8 = {0, 0, 0, 0, 0, 0, 0, 0};
    __builtin_amdgcn_tensor_load_to_lds(g0, g1, gz, gz, gz8, 0);
#else
    __builtin_amdgcn_tensor_load_to_lds(g0, g1, gz, gz, 0);
#endif
}

__global__ void nce_main(const float* __restrict__ x, const float* __restrict__ Zp,
                         double* __restrict__ ws, int B, int numTiles)
{
    extern __shared__ float lds[];                  // 2 buffers x ROWS_PER_TILE x KCOLS floats
    __shared__ double waveAcc[BLOCK_THREADS / 32];

    const int tid  = (int)threadIdx.x;
    const int lane = tid & 31;
    const int wave = tid >> 5;

    const float Z     = Zp[0];
    const float C     = (4096.0f / 1281167.0f) + 1e-7f;  // K1*Pn + EPS
    const float logC1 = __logf(4096.0f / 1281167.0f);    // log(K1*Pn)  (numerator has no EPS)

    const uint32_t ldsBase  = (uint32_t)(uintptr_t)(&lds[0]);   // LDS byte offset of dynamic LDS
    const uint32_t bufBytes = ROWS_PER_TILE * KCOLS * (uint32_t)sizeof(float);

    double acc = 0.0;   // meaningful only on lane 0 of each wave

    // -------- prologue: wave 0 kicks off DMA for this block's first tile --------
    int tile = (int)blockIdx.x;
    if (tid < 32 && tile < numTiles) {
        int rows0 = B - tile * ROWS_PER_TILE;
        rows0 = rows0 > ROWS_PER_TILE ? ROWS_PER_TILE : rows0;
        tdm_load_tile((uint64_t)(uintptr_t)(x + (uint64_t)tile * ROWS_PER_TILE * KP1),
                      ldsBase, (uint32_t)rows0);
    }

    int buf = 0;
    for (; tile < numTiles; tile += (int)gridDim.x, buf ^= 1) {
        const int nextTile = tile + (int)gridDim.x;
        if (tid < 32) {                                  // TDM ignores EXEC: guard whole wave
            if (nextTile < numTiles) {
                int rowsN = B - nextTile * ROWS_PER_TILE;
                rowsN = rowsN > ROWS_PER_TILE ? ROWS_PER_TILE : rowsN;
                tdm_load_tile((uint64_t)(uintptr_t)(x + (uint64_t)nextTile * ROWS_PER_TILE * KP1),
                              ldsBase + (uint32_t)(buf ^ 1) * bufBytes, (uint32_t)rowsN);
                __builtin_amdgcn_s_wait_tensorcnt(1);    // in-order TDM: current tile done
            } else {
                __builtin_amdgcn_s_wait_tensorcnt(0);
            }
        }
        __syncthreads();                                 // publish LDS tile to all waves

        int rowsCur = B - tile * ROWS_PER_TILE;
        rowsCur = rowsCur > ROWS_PER_TILE ? ROWS_PER_TILE : rowsCur;

        if (wave < rowsCur) {
            const float* row = lds + (size_t)buf * ROWS_PER_TILE * KCOLS + (size_t)wave * KCOLS;

            // full-row sum of log(x+C) over k in [0,4096):
            // 4 independent accumulators + unroll 8 keeps ~32 v_log_f32 in flight
            float s0 = 0.0f, s1 = 0.0f, s2 = 0.0f, s3 = 0.0f;
            const float4* rp = (const float4*)row;
            #pragma unroll 8
            for (int i = lane; i < KCOLS / 4; i += 32) {
                float4 v = rp[i];                        // ds_load_b128, conflict-free
                s0 += __logf(v.x + C);
                s1 += __logf(v.y + C);
                s2 += __logf(v.z + C);
                s3 += __logf(v.w + C);
            }
            float s = (s0 + s1) + (s2 + s3);

            // wave32 tree reduction
            #pragma unroll
            for (int off = 16; off >= 1; off >>= 1) s += __shfl_xor(s, off, 32);

            if (lane == 0) {
                const int grow = tile * ROWS_PER_TILE + wave;   // global row index b
                const float x0 = row[0], x1 = row[1], x2 = row[2], x3 = row[3];
                // j = 1 + argmax(x[:,1:5] < MIN_PROB) if any, else 4   (j in [1,4])
                // (x[:,4] < MIN_PROB gives j=4, same as the none-true case)
                const int j = (x1 < MIN_PROB) ? 1 : (x2 < MIN_PROB) ? 2
                            : (x3 < MIN_PROB) ? 3 : 4;

                // lnPon part: sum_{k=j}^{K-1} [log(K1*Pn) - log(x+C)]
                float corr = __logf(x0 + C);                 // k=0 always excluded (j>=1)
                if (j > 1) corr += __logf(x1 + C);
                if (j > 2) corr += __logf(x2 + C);
                if (j > 3) corr += __logf(x3 + C);
                const float lnPon = (float)(KCOLS - j) * logC1 - (s - corr);

                // m-part: m in [0, j)
                float mpart = 0.0f;
                const float xs[4] = {x0, x1, x2, x3};
                #pragma unroll
                for (int m = 0; m < 4; ++m) {
                    if (m < j) {
                        const float xm  = xs[m];
                        const float Pmt = (m == 0) ? xm : xm * (Z / (Z - x0 * Z + xm));
                        const float term = __logf(Pmt) - __logf(Pmt + C);
                        const float w = (grow == 0) ? 1.0f : (float)(j - m);
                        mpart += w * term;
                    }
                }
                acc += (double)(lnPon + mpart);
            }
        }
        __syncthreads();   // everyone done reading buf before TDM may overwrite it
    }

    if (lane == 0) waveAcc[wave] = acc;
    __syncthreads();
    if (tid == 0) {
        double t = 0.0;
        #pragma unroll
        for (int w = 0; w < BLOCK_THREADS / 32; ++w) t += waveAcc[w];
        ws[blockIdx.x] = t;                        // deterministic per-block partial
    }
}

__global__ void nce_final(const double* __restrict__ ws, float* __restrict__ out,
                          int nparts, int B)
{
    if (threadIdx.x == 0 && blockIdx.x == 0) {
        double t = 0.0;
        for (int i = 0; i < nparts; ++i) t += ws[i];   // fixed order: deterministic
        out[0] = (float)(-t / (double)B);
    }
}

extern "C" void kernel_launch(void* const* d_in, const int* in_sizes, int n_in,
                              void* d_out, int out_size, void* d_ws, size_t ws_size,
                              hipStream_t stream)
{
    const float* x  = (const float*)d_in[0];
    // d_in[1] = targets: unused by the reference computation
    const float* Zp = (const float*)d_in[2];
    float* out      = (float*)d_out;
    double* ws      = (double*)d_ws;

    const int B        = in_sizes[0] / KP1;
    const int numTiles = (B + ROWS_PER_TILE - 1) / ROWS_PER_TILE;
    int nblocks        = NBLOCKS;
    if (nblocks > numTiles) nblocks = numTiles;

    // 2 x 64 KB double buffer = 128 KB per block -> 2 blocks co-resident per WGP (320 KB LDS)
    const size_t shmem = 2ull * ROWS_PER_TILE * KCOLS * sizeof(float);

    nce_main<<<nblocks, BLOCK_THREADS, shmem, stream>>>(x, Zp, ws, B, numTiles);
    nce_final<<<1, 32, 0, stream>>>(ws, out, nblocks, B);
}